// L2Net_87342454932162
// MI455X (gfx1250) — compile-verified
//
#include <hip/hip_runtime.h>

typedef float v8f __attribute__((ext_vector_type(8)));
typedef float v2f __attribute__((ext_vector_type(2)));

#define T_STEPS 64
#define NB 32
#define CCH 128
#define DJ 64

__device__ __forceinline__ float wsum32(float v) {
  v += __shfl_xor(v, 16, 32);
  v += __shfl_xor(v, 8, 32);
  v += __shfl_xor(v, 4, 32);
  v += __shfl_xor(v, 2, 32);
  v += __shfl_xor(v, 1, 32);
  return v;
}

// One workgroup per batch element n. 8 waves x 16 channels = 128 channels.
// Each thread holds 32 Jeffress LIF states for one channel, arranged in
// V_WMMA_F32_16X16X4_F32 A-fragment order:
//   lane l: hi = l>>4, m = l&15; chunk kk, vgpr p -> d = 4*kk + 2*hi + p
// B fragment = kint[d] replicated over all 16 N-columns, so the 16x16 f32
// accumulator column n holds zc[channel] identically for every n.
__global__ __launch_bounds__(256)
void snn_scan_kernel(const float* __restrict__ x,
                     const float* __restrict__ w1, const float* __restrict__ b1,
                     const float* __restrict__ w2, const float* __restrict__ b2,
                     const float* __restrict__ sw0, const float* __restrict__ sb0,
                     const float* __restrict__ sw1, const float* __restrict__ sb1,
                     const float* __restrict__ sw2, const float* __restrict__ sb2,
                     float* __restrict__ out)
{
  __shared__ float lds_sw1t[32 * 33];   // sw1 transposed, padded (bank-conflict free)
  __shared__ float lds_part[8];         // per-wave fs partials

  const int n    = blockIdx.x;          // batch element
  const int tid  = threadIdx.x;
  const int lane = tid & 31;
  const int wid  = tid >> 5;
  const int hi   = (lane >> 4) & 1;
  const int m    = lane & 15;
  const int cA   = wid * 16 + m;        // channel whose LIF states live here

  // ---- stage sw1 transposed into LDS ----
  for (int idx = tid; idx < 1024; idx += 256) {
    int i = idx >> 5, j = idx & 31;     // sw1[i][j]
    lds_sw1t[j * 33 + i] = sw1[idx];
  }

  // ---- pack the two binary input spike trains for channel cA into uint64 ----
  unsigned long long bits0 = 0ull, bits1 = 0ull;
  for (int t = 0; t < T_STEPS; ++t) {
    const float* px = x + (((size_t)t * NB + n) * 2) * CCH + cA;
    bits0 |= (unsigned long long)(px[0]   > 0.5f) << t;
    bits1 |= (unsigned long long)(px[CCH] > 0.5f) << t;
  }

  // ---- per-thread kint B-fragment values (exact f32) ----
  v2f kb[16];
  #pragma unroll
  for (int kk = 0; kk < 16; ++kk) {
    #pragma unroll
    for (int p = 0; p < 2; ++p) {
      int d    = 4 * kk + 2 * hi + p;
      int dist = d - DJ / 2;
      int ad   = dist < 0 ? -dist : dist;
      if (ad == 0) ad = 1;              // center clamped to nearest-neighbor weight
      kb[kk][p] = 1.0f / (1.0f - expf(-(float)ad * 0.5f));
    }
  }

  // ---- uniform small weights (scalar loads -> SGPRs) ----
  float w1r[10], b1r[10], w2r[10];
  #pragma unroll
  for (int k = 0; k < 10; ++k) { w1r[k] = w1[k]; b1r[k] = b1[k]; w2r[k] = w2[k]; }
  const float b2s  = b2[0];
  const float sw0r = sw0[lane];
  const float sb0r = sb0[lane];
  const float sb1r = sb1[lane];
  const float sw2r = sw2[lane];
  const float sb2s = sb2[0];

  // ---- persistent state (registers) ----
  float vj[32];
  #pragma unroll
  for (int i = 0; i < 32; ++i) vj[i] = 0.0f;
  float vi = 0.0f, f1 = 0.0f, v2s = 0.0f, fsS = 0.0f;
  float v1[10], f2[10];
  #pragma unroll
  for (int k = 0; k < 10; ++k) { v1[k] = 0.0f; f2[k] = 0.0f; }
  float vs = 0.0f, q2 = 0.0f;                 // per-batch (wave 0)
  float q0v = 0.0f, g1v = 0.0f, q1v = 0.0f, g2v = 0.0f;

  unsigned long long h0 = 0ull, h1 = 0ull;    // delay-line history registers

  __syncthreads();

  #pragma unroll 1
  for (int t = 0; t < T_STEPS; ++t) {
    h0 = (h0 << 1) | ((bits0 >> t) & 1ull);   // bit j == x0[t-j]
    h1 = (h1 << 1) | ((bits1 >> t) & 1ull);   // bit j == x1[t-j]

    // ---- Jeffress LIF update + WMMA kint reduction ----
    v8f acc = {0.f, 0.f, 0.f, 0.f, 0.f, 0.f, 0.f, 0.f};
    #pragma unroll
    for (int kk = 0; kk < 16; ++kk) {
      v2f a;
      #pragma unroll
      for (int p = 0; p < 2; ++p) {
        int d = 4 * kk + 2 * hi + p;
        float u = (float)(((h0 >> d) & 1ull) + ((h1 >> (63 - d)) & 1ull));
        float v = vj[2 * kk + p];
        v += (u - v) * (1.0f / 20.0f);        // LIF tau = 20, decay_input
        float s = (v >= 1.0f) ? 1.0f : 0.0f;  // fire
        vj[2 * kk + p] = (1.0f - s) * v;      // hard reset
        a[p] = s;
      }
      // D = A(spikes 16x4) * B(kint 4x16, column-replicated) + C
      acc = __builtin_amdgcn_wmma_f32_16x16x4_f32(false, a, false, kb[kk],
                                                  (short)0, acc, false, false);
    }

    // acc[r] (this lane) = zc for channel wid*16 + r + 8*hi, any column.
    // Owned channel = wid*16 + (lane&7) + 8*hi  (lanes l and l+8 duplicate).
    const int r = lane & 7;
    float zc = acc[0];
    #pragma unroll
    for (int rr = 1; rr < 8; ++rr) zc = (r == rr) ? acc[rr] : zc;

    // ---- per-channel scalar chain ----
    vi += zc;
    float s2 = (vi >= 1.0f) ? 1.0f : 0.0f; vi *= (1.0f - s2);
    f1 = f1 * 0.5f + s2;                      // f - f/2 is exact for tau=2
    float v2add = 0.0f;
    #pragma unroll
    for (int k = 0; k < 10; ++k) {
      v1[k] += f1 * w1r[k] + b1r[k];
      float s3 = (v1[k] >= 1.0f) ? 1.0f : 0.0f;
      v1[k] *= (1.0f - s3);
      f2[k] = f2[k] * 0.5f + s3;
      v2add += f2[k] * w2r[k];
    }
    v2s += v2add + b2s;
    float s4 = (v2s >= 1.0f) ? 1.0f : 0.0f; v2s *= (1.0f - s4);
    fsS = fsS * 0.5f + s4;

    // ---- sum fs over the wave's 16 channels (mask out duplicates) ----
    float contrib = ((lane & 8) == 0) ? fsS : 0.0f;
    contrib = wsum32(contrib);
    if (lane == 0) lds_part[wid] = contrib;
    __syncthreads();

    // ---- per-batch tail: sum IF + sqrt-model MLP (wave 0 only) ----
    if (wid == 0) {
      float pv = (lane < 8) ? lds_part[lane] : 0.0f;
      float fstot = wsum32(pv);
      vs += fstot;
      float s5 = (vs >= 1.0f) ? 1.0f : 0.0f; vs *= (1.0f - s5);
      float h = s5;
      q0v += h * sw0r + sb0r;
      float s6 = (q0v >= 1.0f) ? 1.0f : 0.0f; q0v *= (1.0f - s6);
      g1v = g1v * 0.5f + s6;
      float acc1 = 0.0f;
      #pragma unroll 8
      for (int j = 0; j < 32; ++j) {
        float gj = __shfl(g1v, j, 32);
        acc1 += gj * lds_sw1t[j * 33 + lane];
      }
      q1v += acc1 + sb1r;
      float s7 = (q1v >= 1.0f) ? 1.0f : 0.0f; q1v *= (1.0f - s7);
      g2v = g2v * 0.5f + s7;
      float dot = wsum32(g2v * sw2r);
      q2 += dot + sb2s;                       // NonSpikingIF: output = v
      if (lane == 0) out[t * NB + n] = q2;
    }
    __syncthreads();                          // protect lds_part for next step
  }
}

extern "C" void kernel_launch(void* const* d_in, const int* in_sizes, int n_in,
                              void* d_out, int out_size, void* d_ws, size_t ws_size,
                              hipStream_t stream) {
  const float* x   = (const float*)d_in[0];
  const float* w1  = (const float*)d_in[1];
  const float* b1  = (const float*)d_in[2];
  const float* w2  = (const float*)d_in[3];
  const float* b2  = (const float*)d_in[4];
  const float* sw0 = (const float*)d_in[5];
  const float* sb0 = (const float*)d_in[6];
  const float* sw1 = (const float*)d_in[7];
  const float* sb1 = (const float*)d_in[8];
  const float* sw2 = (const float*)d_in[9];
  const float* sb2 = (const float*)d_in[10];
  float* out = (float*)d_out;

  snn_scan_kernel<<<dim3(NB), dim3(256), 0, stream>>>(
      x, w1, b1, w2, b2, sw0, sb0, sw1, sb1, sw2, sb2, out);
}